// Top1Gate_11940009083382
// MI455X (gfx1250) — compile-verified
//
#include <hip/hip_runtime.h>

#define NUM_EXPERTS  64
#define MODEL_DIM    4096
#define NUM_TOKENS   16384
#define BLOCK_TOKENS 64
#define KSPLIT       2
#define KCHUNK       (MODEL_DIM / KSPLIT)     // 2048 per wave
#define WAVES_PER_BLOCK 8                     // 4 token-tiles x 2 K-split waves
#define THREADS (WAVES_PER_BLOCK * 32)

typedef __attribute__((ext_vector_type(2))) float v2f;
typedef __attribute__((ext_vector_type(8))) float v8f;

// Wave pair (same token tile, K split in half) each computes a 16-token x 64-expert
// partial logit tile with V_WMMA_F32_16X16X4_F32; partials are combined through LDS,
// then the even wave does softmax + top-1 via shfl_xor butterflies.
__global__ __launch_bounds__(THREADS) void top1gate_kernel(
    const float* __restrict__ x, const float* __restrict__ W,
    float* __restrict__ out, float* __restrict__ ws_me, float* __restrict__ ws_ce)
{
    __shared__ float slab[NUM_EXPERTS * BLOCK_TOKENS];    // combine slab [E][64 tokens]
    __shared__ v8f   part[WAVES_PER_BLOCK / 2][32][4];    // odd-wave partial accumulators
    __shared__ float lds_me[NUM_EXPERTS];
    __shared__ float lds_ce[NUM_EXPERTS];

    const int tid  = threadIdx.x;
    const int lane = tid & 31;
    const int wave = tid >> 5;
    const int pair = wave >> 1;          // which 16-token tile
    const int kow  = (wave & 1);         // which K half
    const int col  = lane & 15;          // A: token row M ; B: expert col N
    const int half = lane >> 4;          // K half select per ISA 16x4 f32 layout
    const int khalf = half << 1;         // k offset {0,2}

    // zero shared accumulators + combine slab
    if (tid < NUM_EXPERTS)            lds_me[tid] = 0.0f;
    else if (tid < 2 * NUM_EXPERTS)   lds_ce[tid - NUM_EXPERTS] = 0.0f;
    for (int i = tid; i < NUM_EXPERTS * BLOCK_TOKENS; i += THREADS) slab[i] = 0.0f;
    __syncthreads();

    const int tokBase = blockIdx.x * BLOCK_TOKENS + pair * 16;
    const int kOff    = kow * KCHUNK;

    // A fragment: x[(tokBase+col)*D + k + 2*half .. +1]   (contiguous pair)
    // B fragment: W[(ntile*16+col)*D + k + 2*half .. +1]  (contiguous pair)
    const float* pa  = x + (size_t)(tokBase + col) * MODEL_DIM + kOff + khalf;
    const float* pb0 = W + (size_t)col * MODEL_DIM + kOff + khalf;
    const float* pb1 = pb0 + (size_t)16 * MODEL_DIM;
    const float* pb2 = pb0 + (size_t)32 * MODEL_DIM;
    const float* pb3 = pb0 + (size_t)48 * MODEL_DIM;

    v8f acc0 = {}, acc1 = {}, acc2 = {}, acc3 = {};

    #pragma unroll 8
    for (int k = 0; k < KCHUNK; k += 4) {
        v2f A  = __builtin_nontemporal_load((const v2f*)(pa  + k)); // x: read-once, stream
        v2f B0 = *(const v2f*)(pb0 + k);                            // W: cache-resident
        v2f B1 = *(const v2f*)(pb1 + k);
        v2f B2 = *(const v2f*)(pb2 + k);
        v2f B3 = *(const v2f*)(pb3 + k);
        acc0 = __builtin_amdgcn_wmma_f32_16x16x4_f32(false, A, false, B0, (short)0, acc0, false, false);
        acc1 = __builtin_amdgcn_wmma_f32_16x16x4_f32(false, A, false, B1, (short)0, acc1, false, false);
        acc2 = __builtin_amdgcn_wmma_f32_16x16x4_f32(false, A, false, B2, (short)0, acc2, false, false);
        acc3 = __builtin_amdgcn_wmma_f32_16x16x4_f32(false, A, false, B3, (short)0, acc3, false, false);
    }

    // K-split reduction: odd wave publishes partials, even wave folds them in.
    if (kow == 1) {
        part[pair][lane][0] = acc0;
        part[pair][lane][1] = acc1;
        part[pair][lane][2] = acc2;
        part[pair][lane][3] = acc3;
    }
    __syncthreads();

    if (kow == 0) {
        acc0 += part[pair][lane][0];
        acc1 += part[pair][lane][1];
        acc2 += part[pair][lane][2];
        acc3 += part[pair][lane][3];

        // C/D layout: VGPR r holds token M=r (lanes 0-15) / M=r+8 (lanes 16-31); N = col.
        // Each token's 64 logits live in one 16-lane half across the 4 acc tiles.
        float meAcc0 = 0.f, meAcc1 = 0.f, meAcc2 = 0.f, meAcc3 = 0.f;

        #pragma unroll
        for (int r = 0; r < 8; ++r) {
            float l0 = acc0[r], l1 = acc1[r], l2 = acc2[r], l3 = acc3[r];

            // softmax max over 64 experts (4 local + butterfly over 16 lanes in half)
            float m = fmaxf(fmaxf(l0, l1), fmaxf(l2, l3));
            #pragma unroll
            for (int mk = 1; mk <= 8; mk <<= 1)
                m = fmaxf(m, __shfl_xor(m, mk, 32));

            float e0 = __expf(l0 - m);
            float e1 = __expf(l1 - m);
            float e2 = __expf(l2 - m);
            float e3 = __expf(l3 - m);
            float s = (e0 + e1) + (e2 + e3);
            #pragma unroll
            for (int mk = 1; mk <= 8; mk <<= 1)
                s += __shfl_xor(s, mk, 32);
            float inv = 1.0f / s;
            float g0 = e0 * inv, g1 = e1 * inv, g2 = e2 * inv, g3 = e3 * inv;

            meAcc0 += g0; meAcc1 += g1; meAcc2 += g2; meAcc3 += g3;

            // top-1 (first-max tie-break, like jnp.argmax)
            float bv = g0; int bi = col;
            if (g1 > bv) { bv = g1; bi = col + 16; }
            if (g2 > bv) { bv = g2; bi = col + 32; }
            if (g3 > bv) { bv = g3; bi = col + 48; }
            #pragma unroll
            for (int mk = 1; mk <= 8; mk <<= 1) {
                float ov = __shfl_xor(bv, mk, 32);
                int   oi = __shfl_xor(bi, mk, 32);
                if (ov > bv || (ov == bv && oi < bi)) { bv = ov; bi = oi; }
            }

            if (col == 0) {  // one writer per (r, half) => per token
                int tl = pair * 16 + r + (half << 3);
                slab[bi * BLOCK_TOKENS + tl] = bv;    // combine[top1][token] = p
                atomicAdd(&lds_ce[bi], 1.0f);         // top-1 count
            }
        }

        // fold the two halves' token sets, then accumulate per-expert gate sums
        meAcc0 += __shfl_xor(meAcc0, 16, 32);
        meAcc1 += __shfl_xor(meAcc1, 16, 32);
        meAcc2 += __shfl_xor(meAcc2, 16, 32);
        meAcc3 += __shfl_xor(meAcc3, 16, 32);
        if (half == 0) {
            atomicAdd(&lds_me[col],      meAcc0);
            atomicAdd(&lds_me[col + 16], meAcc1);
            atomicAdd(&lds_me[col + 32], meAcc2);
            atomicAdd(&lds_me[col + 48], meAcc3);
        }
    }
    __syncthreads();

    // block-level partials -> global workspace
    if (tid < NUM_EXPERTS)           atomicAdd(&ws_me[tid], lds_me[tid]);
    else if (tid < 2 * NUM_EXPERTS)  atomicAdd(&ws_ce[tid - NUM_EXPERTS], lds_ce[tid - NUM_EXPERTS]);

    // coalesced store of combine slab: out layout [E, T], block owns 64 token cols
    float* comb = out + 1;  // d_out[0] = l_aux
    const size_t colBase = (size_t)blockIdx.x * BLOCK_TOKENS;
    for (int i = tid; i < NUM_EXPERTS * BLOCK_TOKENS; i += THREADS) {
        int e  = i >> 6;
        int tl = i & (BLOCK_TOKENS - 1);
        comb[(size_t)e * NUM_TOKENS + colBase + tl] = slab[i];
    }
}

__global__ __launch_bounds__(NUM_EXPERTS) void finalize_kernel(
    const float* __restrict__ ws, float* __restrict__ out)
{
    __shared__ float red[NUM_EXPERTS];
    int t = threadIdx.x;
    // me = ws[t]/T, ce = ws[64+t]/T ; l_aux = mean(me*ce)*E^2 = E * sum(me_sum*ce_cnt)/T^2
    red[t] = ws[t] * ws[t + NUM_EXPERTS];
    __syncthreads();
    for (int s = NUM_EXPERTS / 2; s > 0; s >>= 1) {
        if (t < s) red[t] += red[t + s];
        __syncthreads();
    }
    if (t == 0) {
        const float T = (float)NUM_TOKENS;
        out[0] = red[0] * (float)NUM_EXPERTS / (T * T);
    }
}

extern "C" void kernel_launch(void* const* d_in, const int* in_sizes, int n_in,
                              void* d_out, int out_size, void* d_ws, size_t ws_size,
                              hipStream_t stream) {
    (void)in_sizes; (void)n_in; (void)out_size; (void)ws_size;
    const float* x = (const float*)d_in[0];
    const float* W = (const float*)d_in[1];
    float* out = (float*)d_out;
    float* ws  = (float*)d_ws;

    hipMemsetAsync(d_ws, 0, 2 * NUM_EXPERTS * sizeof(float), stream);

    dim3 grid(NUM_TOKENS / BLOCK_TOKENS);
    top1gate_kernel<<<grid, THREADS, 0, stream>>>(x, W, out, ws, ws + NUM_EXPERTS);
    finalize_kernel<<<1, NUM_EXPERTS, 0, stream>>>(ws, out);
}